// RunningStats_85839216378453
// MI455X (gfx1250) — compile-verified
//
#include <hip/hip_runtime.h>

// ---------------------------------------------------------------------------
// RunningStats (per-channel standardize) for MI455X / gfx1250.
// x: [B=32, C=64, H=256, W=256] f32, channels-first.
//
// Roofline: ~4 flops vs 12 bytes per element -> purely HBM-bound.
//   pass1 read 512 MiB + pass3 read 512 MiB + write 512 MiB = 1.5 GiB
//   @ 23.3 TB/s  =>  ~66 us floor. 512 MiB >> 192 MB L2 -> NT hints.
//
// Pass 1: per-(plane,chunk) partial sum/sumsq. Each wave stages its slab
//         through LDS with 8 outstanding GLOBAL_LOAD_ASYNC_TO_LDS_B128 ops
//         (ASYNCcnt), drained with descending s_wait_asynccnt immediates.
//         No workgroup barriers needed: lanes consume only their own bytes,
//         and ASYNCcnt is per-wave. No LDS buffer reuse -> no WAR hazard.
// Pass 2: per-channel deterministic tree reduce -> mean, inv_std.
// Pass 3: streaming normalize, non-temporal b128 in / b128 out.
// ---------------------------------------------------------------------------

typedef float f4 __attribute__((ext_vector_type(4)));

#define EPS_F 1e-8f

constexpr int Bn = 32;
constexpr int Cn = 64;
constexpr int Hn = 256;
constexpr int Wn = 256;
constexpr int PLANE = Hn * Wn;                 // 65536 contiguous floats per (b,c)
constexpr int CHUNKS = 8;                      // blocks per plane
constexpr int CHUNK_ELEMS = PLANE / CHUNKS;    // 8192 floats per block
constexpr int TPB = 256;                       // 8 wave32 per block
constexpr int TILE_ELEMS = TPB * 4;            // 1024 floats = 4 KiB per tile
constexpr int TILES = CHUNK_ELEMS / TILE_ELEMS;// 8 tiles, all resident in LDS
constexpr int PARTIALS = Bn * CHUNKS;          // 256 partials per channel
constexpr float NF = (float)((long long)Bn * PLANE); // 2097152

// --------------------------- gfx1250 async path ----------------------------
#if defined(__has_builtin)
#  if __has_builtin(__builtin_amdgcn_global_load_async_to_lds_b128) && \
      __has_builtin(__builtin_amdgcn_s_wait_asynccnt)
#    define USE_ASYNC_LDS 1
#  endif
#endif
#ifndef USE_ASYNC_LDS
#  define USE_ASYNC_LDS 0
#endif

#if USE_ASYNC_LDS
// ROCm 7.2 clang-22 signature (probed): typed pointers to 16-byte int vector.
typedef int vi4 __attribute__((vector_size(16)));
typedef __attribute__((address_space(1))) vi4* g_b128_ptr;
typedef __attribute__((address_space(3))) vi4* l_b128_ptr;

__device__ __forceinline__ void async_copy_b128(void* lds_dst, const void* gsrc) {
  // GLOBAL_LOAD_ASYNC_TO_LDS_B128: tracked by per-wave ASYNCcnt.
  __builtin_amdgcn_global_load_async_to_lds_b128(
      (g_b128_ptr)gsrc,
      (l_b128_ptr)lds_dst,
      /*offset=*/0, /*cpol=*/0);
}
#endif

// ------------------------- pass 1: partial reduce --------------------------
__global__ __launch_bounds__(TPB) void k_partial(const float* __restrict__ x,
                                                 float* __restrict__ psum,
                                                 float* __restrict__ psq) {
  const int tid   = threadIdx.x;
  const int blk   = blockIdx.x;          // [0, 16384)
  const int plane = blk >> 3;            // b*64 + c
  const int chunk = blk & 7;
  const int c     = plane & (Cn - 1);
  const int b     = plane >> 6;
  const float* base =
      x + (size_t)plane * PLANE + (size_t)chunk * CHUNK_ELEMS;

  float sum = 0.0f, sq = 0.0f;

#if USE_ASYNC_LDS
  __shared__ __align__(16) float tile[TILES][TILE_ELEMS];   // 32 KiB, no reuse

  // Issue all TILES async b128 loads up-front: 8-deep memory-level
  // parallelism per wave, completion returns in order (ASYNCcnt).
#pragma unroll
  for (int t = 0; t < TILES; ++t) {
    async_copy_b128(&tile[t][tid * 4],
                    base + (size_t)t * TILE_ELEMS + tid * 4);
  }

  // Drain in order. Each lane reads only bytes its own async load wrote,
  // so a per-wave asynccnt wait is sufficient -- no workgroup barrier.
  // s_wait_asynccnt requires an immediate => macro-unrolled constants.
#define CONSUME_TILE(t)                                                    \
  {                                                                        \
    __builtin_amdgcn_s_wait_asynccnt(TILES - 1 - (t));                     \
    const f4 v = *(const f4*)&tile[(t)][tid * 4];                          \
    sum += (v.x + v.y) + (v.z + v.w);                                      \
    sq = fmaf(v.x, v.x, fmaf(v.y, v.y, fmaf(v.z, v.z, fmaf(v.w, v.w, sq))));\
  }
  CONSUME_TILE(0)
  CONSUME_TILE(1)
  CONSUME_TILE(2)
  CONSUME_TILE(3)
  CONSUME_TILE(4)
  CONSUME_TILE(5)
  CONSUME_TILE(6)
  CONSUME_TILE(7)
#undef CONSUME_TILE
#else
  for (int t = 0; t < TILES; ++t) {
    const f4 v = __builtin_nontemporal_load(
        (const f4*)(base + (size_t)t * TILE_ELEMS + tid * 4));
    sum += (v.x + v.y) + (v.z + v.w);
    sq = fmaf(v.x, v.x, fmaf(v.y, v.y, fmaf(v.z, v.z, fmaf(v.w, v.w, sq))));
  }
#endif

  // deterministic block tree-reduce (only place barriers are needed)
  __shared__ float r0[TPB];
  __shared__ float r1[TPB];
  r0[tid] = sum;
  r1[tid] = sq;
  __syncthreads();
  for (int s = TPB / 2; s > 0; s >>= 1) {
    if (tid < s) {
      r0[tid] += r0[tid + s];
      r1[tid] += r1[tid + s];
    }
    __syncthreads();
  }
  if (tid == 0) {
    const int p = c * PARTIALS + b * CHUNKS + chunk;
    psum[p] = r0[0];
    psq[p]  = r1[0];
  }
}

// ----------------------- pass 2: finalize statistics -----------------------
__global__ __launch_bounds__(PARTIALS) void k_finalize(
    const float* __restrict__ psum, const float* __restrict__ psq,
    float* __restrict__ mean, float* __restrict__ istd) {
  __shared__ float r0[PARTIALS];
  __shared__ float r1[PARTIALS];
  const int c = blockIdx.x;
  const int tid = threadIdx.x;
  r0[tid] = psum[c * PARTIALS + tid];
  r1[tid] = psq[c * PARTIALS + tid];
  __syncthreads();
  for (int s = PARTIALS / 2; s > 0; s >>= 1) {
    if (tid < s) {
      r0[tid] += r0[tid + s];
      r1[tid] += r1[tid + s];
    }
    __syncthreads();
  }
  if (tid == 0) {
    const float m   = r0[0] * (1.0f / NF);
    const float m2  = fmaf(-NF * m, m, r1[0]);          // sumsq - n*mean^2
    const float var = fmaxf(m2 * (1.0f / (NF - 1.0f)), EPS_F);
    mean[c] = m;
    istd[c] = rsqrtf(var + EPS_F);
  }
}

// ----------------------- pass 3: streaming normalize -----------------------
__global__ __launch_bounds__(TPB) void k_standardize(
    const float* __restrict__ x, const float* __restrict__ mean,
    const float* __restrict__ istd, float* __restrict__ out) {
  const size_t i = (size_t)blockIdx.x * TPB + threadIdx.x;
  const size_t e = i * 4;                         // 4 consecutive w -> same c
  const int c = (int)((e >> 16) & (Cn - 1));      // PLANE == 2^16

  const f4 v = __builtin_nontemporal_load((const f4*)(x + e));
  const float m = mean[c];
  const float s = istd[c];
  f4 r;
  r.x = (v.x - m) * s;
  r.y = (v.y - m) * s;
  r.z = (v.z - m) * s;
  r.w = (v.w - m) * s;
  __builtin_nontemporal_store(r, (f4*)(out + e));
}

// ---------------------------------------------------------------------------
extern "C" void kernel_launch(void* const* d_in, const int* in_sizes, int n_in,
                              void* d_out, int out_size, void* d_ws,
                              size_t ws_size, hipStream_t stream) {
  (void)in_sizes; (void)n_in; (void)out_size; (void)ws_size;

  const float* x = (const float*)d_in[0];
  float* out = (float*)d_out;

  // workspace layout (all fully rewritten every call -> deterministic):
  //   psum[Cn*PARTIALS] | psq[Cn*PARTIALS] | mean[Cn] | istd[Cn]
  float* ws   = (float*)d_ws;
  float* psum = ws;
  float* psq  = ws + (size_t)Cn * PARTIALS;
  float* mean = ws + 2 * (size_t)Cn * PARTIALS;
  float* istd = mean + Cn;

  const int grid1 = Bn * Cn * CHUNKS;                          // 16384
  const int grid3 = (Bn * Cn * PLANE) / 4 / TPB;               // 131072

  k_partial<<<grid1, TPB, 0, stream>>>(x, psum, psq);
  k_finalize<<<Cn, PARTIALS, 0, stream>>>(psum, psq, mean, istd);
  k_standardize<<<grid3, TPB, 0, stream>>>(x, mean, istd, out);
}